// HierarchicalSoftmaxLoss_76373108457493
// MI455X (gfx1250) — compile-verified
//
#include <hip/hip_runtime.h>

// Hierarchical softmax NLL, MI455X (gfx1250, wave32).
//
// Tree walk is data-independent: node_k = (2^k - 1) + (class_idx >> (16 - k)),
// so all 16 gather addresses are computed up front and all 16 global_load_b32
// are issued as one clause -> one memory latency, not a 16-deep chain.
//
// Per-step loss: bit=1 (right): -log(1-sigmoid(s)) = softplus(+s)
//                bit=0 (left) : -log(sigmoid(s))   = softplus(-s)
// softplus(x) = max(x,0) + log(1 + exp(-|x|)), evaluated with the gfx1250
// hardware transcendentals (v_exp_f32 / v_log_f32 via base-2 builtins):
// the log argument is 1+t with t in (0,1]; when t is tiny the max(x,0) term
// dominates, so plain log of (1+t) is accurate enough (abs err ~1e-7).

#define HSM_CODE_LEN 16
#define HSM_V        65536

#define LOG2E_F 1.44269504088896340736f
#define LN2_F   0.69314718055994530942f

__global__ __launch_bounds__(256) void
hsm_partial_kernel(const float* __restrict__ scores,
                   const int*   __restrict__ cls,
                   float*       __restrict__ partial,  // one per block
                   int B)
{
    __shared__ float s_wave[8];  // 256 threads = 8 wave32 waves

    const int b = blockIdx.x * 256 + threadIdx.x;

    float acc = 0.0f;
    if (b < B) {
        const int c = cls[b];
        const float* __restrict__ row = scores + (size_t)b * (size_t)HSM_V;

        // Phase 1: pure-ALU address generation, then a clustered 16-load
        // gather clause (addresses are independent of loaded data).
        float s[HSM_CODE_LEN];
#pragma unroll
        for (int k = 0; k < HSM_CODE_LEN; ++k) {
            const int node = ((1 << k) - 1) +
                             (int)((unsigned)c >> (HSM_CODE_LEN - k));
            s[k] = __builtin_nontemporal_load(row + node);  // th:NT, no reuse
        }

        // Phase 2: softplus accumulation on hardware transcendentals.
        float acc2 = 0.0f;   // sum of log2(1+exp(-|x|)) terms (base-2 domain)
#pragma unroll
        for (int k = 0; k < HSM_CODE_LEN; ++k) {
            const bool right = ((c >> (HSM_CODE_LEN - 1 - k)) & 1) != 0;
            const float x  = right ? s[k] : -s[k];
            const float ax = __builtin_fabsf(x);
            const float t  = __builtin_amdgcn_exp2f(-ax * LOG2E_F); // exp(-|x|)
            acc2 += __builtin_amdgcn_logf(1.0f + t);                // log2(1+t)
            acc  += fmaxf(x, 0.0f);
        }
        acc = __builtin_fmaf(acc2, LN2_F, acc);
    }

    // wave32 butterfly reduction (lane-permute path).
#pragma unroll
    for (int off = 16; off > 0; off >>= 1)
        acc += __shfl_xor(acc, off, 32);

    const int lane = threadIdx.x & 31;
    const int wave = threadIdx.x >> 5;
    if (lane == 0) s_wave[wave] = acc;
    __syncthreads();

    // Deterministic fixed-order fold of the 8 wave partials (no atomics ->
    // bit-stable across graph replays, no d_out pre-zero needed).
    if (threadIdx.x == 0) {
        float t = 0.0f;
#pragma unroll
        for (int w = 0; w < 8; ++w) t += s_wave[w];
        partial[blockIdx.x] = t;
    }
}

__global__ void hsm_final_kernel(const float* __restrict__ partial,
                                 float* __restrict__ out,
                                 int nPartials, float invB)
{
    if (threadIdx.x == 0 && blockIdx.x == 0) {
        float t = 0.0f;
        for (int i = 0; i < nPartials; ++i) t += partial[i];  // fixed order
        out[0] = t * invB;
    }
}

extern "C" void kernel_launch(void* const* d_in, const int* in_sizes, int n_in,
                              void* d_out, int out_size, void* d_ws, size_t ws_size,
                              hipStream_t stream)
{
    const float* scores = (const float*)d_in[0];
    const int*   cls    = (const int*)d_in[1];   // integer input -> const int*
    float*       out    = (float*)d_out;

    const int B = in_sizes[1];                   // 2048 rows
    const int nBlocks = (B + 255) / 256;         // 8 blocks of 8 wave32 waves

    float* partial = (float*)d_ws;               // nBlocks floats of scratch

    hsm_partial_kernel<<<nBlocks, 256, 0, stream>>>(scores, cls, partial, B);
    hsm_final_kernel<<<1, 32, 0, stream>>>(partial, out, nBlocks, 1.0f / (float)B);

    (void)n_in; (void)out_size; (void)ws_size;
}